// Model_63230508532151
// MI455X (gfx1250) — compile-verified
//
#include <hip/hip_runtime.h>

typedef __attribute__((ext_vector_type(2))) float v2f;
typedef __attribute__((ext_vector_type(8))) float v8f;

#define Hn 1024
#define Wn 1024
#define Cn 3
#define GHn 16
#define GWn 16
#define GD 8
#define NC 12
#define TH 16
#define TW 32
#define NTHREADS 128

#if defined(__gfx1250__) && __has_builtin(__builtin_amdgcn_global_load_async_to_lds_b32)
#define ASYNC_STAGE 1
#endif

// fused: 3x3 SAME conv (+bias) -> bilateral grid slice+apply, one HBM pass
__global__ __launch_bounds__(NTHREADS)
void fused_conv_bilateral(const float* __restrict__ bgrid,
                          const float* __restrict__ guide,
                          const float* __restrict__ image,
                          const float* __restrict__ conv_w,
                          const float* __restrict__ conv_b,
                          float* __restrict__ out)
{
    __shared__ float lds_in[(TH + 2) * (TW + 2) * Cn];   // padded conv input tile
    __shared__ float lds_x[TH * TW * Cn];                // conv output tile
    __shared__ float lds_grid[3 * 3 * GD * NC];          // grid sub-block (edge-clipped)

    const int tid   = threadIdx.x;
    const int tileX = blockIdx.x, tileY = blockIdx.y, batch = blockIdx.z;
    const int x0 = tileX * TW, y0 = tileY * TH;

    // base grid cells covered by this tile (64 px per cell in x and y)
    const int fy0 = (int)floorf((y0 + 0.5f) * (1.0f / 64.0f) - 0.5f);
    const int fx0 = (int)floorf((x0 + 0.5f) * (1.0f / 64.0f) - 0.5f);

    // ---- stage image tile (zero-padded halo) into LDS ----
    for (int idx = tid; idx < (TH + 2) * (TW + 2) * Cn; idx += NTHREADS) {
        int c  = idx % Cn;
        int t  = idx / Cn;
        int lx = t % (TW + 2);
        int ly = t / (TW + 2);
        int gy = y0 + ly - 1, gx = x0 + lx - 1;
        if (gy >= 0 && gy < Hn && gx >= 0 && gx < Wn) {
            const float* src = &image[(((size_t)batch * Hn + gy) * Wn + gx) * Cn + c];
#ifdef ASYNC_STAGE
            // direct global->LDS async copy (ASYNCcnt), no VGPR round trip
            __builtin_amdgcn_global_load_async_to_lds_b32(
                (__attribute__((address_space(1))) int*)src,
                (__attribute__((address_space(3))) int*)&lds_in[idx],
                0, 0);
#else
            lds_in[idx] = *src;
#endif
        } else {
            lds_in[idx] = 0.0f;   // zero padding (disjoint from async targets)
        }
    }
    // ---- stage 3x3 (y,x) cells x 8 z x 12 ch of the bilateral grid ----
    for (int idx = tid; idx < 3 * 3 * GD * NC; idx += NTHREADS) {
        int c  = idx % NC;
        int z  = (idx / NC) % GD;
        int jx = (idx / (NC * GD)) % 3;
        int jy = idx / (NC * GD * 3);
        int cy = min(max(fy0 + jy, 0), GHn - 1);
        int cx = min(max(fx0 + jx, 0), GWn - 1);
        lds_grid[idx] = bgrid[((((size_t)batch * GHn + cy) * GWn + cx) * GD + z) * NC + c];
    }
    // prefetch guide tile for the slice phase (global_prefetch_b8)
    {
        const float* gptr = &guide[((size_t)batch * Hn + y0 + (tid >> 5) * 4) * Wn + x0];
        __builtin_prefetch(gptr, 0, 0);
    }
#ifdef ASYNC_STAGE
#if __has_builtin(__builtin_amdgcn_s_wait_asynccnt)
    __builtin_amdgcn_s_wait_asynccnt(0);
#else
    asm volatile("s_wait_asynccnt 0x0" ::: "memory");
#endif
#endif
    __syncthreads();

    const int wave = tid >> 5;
    const int lane = tid & 31;
    const int half = lane >> 4;   // lane group: K pair select
    const int n    = lane & 15;   // B/D column (output channel), A row (pixel)

    // ---- B fragments: weights as 28x16 (K x N); K = tap*3 + cin, K==27 -> bias ----
    v2f bfr[7];
#pragma unroll
    for (int s = 0; s < 7; ++s) {
        float bv0 = 0.0f, bv1 = 0.0f;
        if (n < Cn) {
            int k0 = 4 * s + 2 * half;
            // conv_w layout (ky,kx,ci,co): flat = ((ky*3+kx)*3+ci)*3 + co = k*3 + co
            bv0 = (k0     < 27) ? conv_w[(k0)     * Cn + n] : conv_b[n];
            bv1 = (k0 + 1 < 27) ? conv_w[(k0 + 1) * Cn + n] : conv_b[n];
        }
        bfr[s].x = bv0;
        bfr[s].y = bv1;
    }

    // ---- conv via V_WMMA_F32_16X16X4_F32: M=16 px, N=3(of 16) ch, K=28 in 7 steps ----
    for (int rr = 0; rr < 4; ++rr) {
        const int row = wave * 4 + rr;           // local row in tile
        for (int g = 0; g < 2; ++g) {
            const int px = g * 16 + n;           // this lane's A-row pixel column
            v8f acc = {};
#pragma unroll
            for (int s = 0; s < 7; ++s) {
                const int k0 = 4 * s + 2 * half;
                float av[2];
#pragma unroll
                for (int v = 0; v < 2; ++v) {
                    int k = k0 + v;
                    float val = 1.0f;            // K==27: bias lane (A = 1.0)
                    if (k < 27) {
                        int tap = k / 3, ci = k - tap * 3;
                        int ky = tap / 3, kx = tap - ky * 3;
                        val = lds_in[((row + ky) * (TW + 2) + (px + kx)) * Cn + ci];
                    }
                    av[v] = val;
                }
                v2f a; a.x = av[0]; a.y = av[1];
                acc = __builtin_amdgcn_wmma_f32_16x16x4_f32(
                    false, a, false, bfr[s], (short)0, acc, false, false);
            }
            // scatter D (col = n, rows M = r + 8*half) to pixel-major LDS
            if (n < Cn) {
#pragma unroll
                for (int r = 0; r < 8; ++r) {
                    int m = r + 8 * half;
                    lds_x[(row * TW + g * 16 + m) * Cn + n] = acc[r];
                }
            }
        }
    }
    __syncthreads();

    // ---- bilateral slice + affine apply, 4 px per thread ----
    for (int i = 0; i < 4; ++i) {
        int p   = i * NTHREADS + tid;
        int row = p >> 5;          // / TW
        int col = p & 31;          // % TW
        int yg = y0 + row, xg = x0 + col;

        float gxf = (xg + 0.5f) * (1.0f / 64.0f) - 0.5f;
        float gyf = (yg + 0.5f) * (1.0f / 64.0f) - 0.5f;
        float gv  = guide[((size_t)batch * Hn + yg) * Wn + xg];
        float gzf = gv * (float)GD - 0.5f;

        float fxf = floorf(gxf), fyf = floorf(gyf), fzf = floorf(gzf);
        float wx1 = gxf - fxf, wx0v = 1.0f - wx1;
        float wy1 = gyf - fyf, wy0v = 1.0f - wy1;
        float wz1 = gzf - fzf, wz0v = 1.0f - wz1;
        int jx = (int)fxf - fx0;
        int jy = (int)fyf - fy0;
        int zi0 = min(max((int)fzf,     0), GD - 1);
        int zi1 = min(max((int)fzf + 1, 0), GD - 1);

        const float* xr = &lds_x[(row * TW + col) * Cn];
        float xv0 = xr[0], xv1 = xr[1], xv2 = xr[2];

        float o0 = 0.0f, o1 = 0.0f, o2 = 0.0f;
#pragma unroll
        for (int dz = 0; dz < 2; ++dz) {
            float wz = dz ? wz1 : wz0v;
            int   zi = dz ? zi1 : zi0;
#pragma unroll
            for (int dy = 0; dy < 2; ++dy) {
                float wy = dy ? wy1 : wy0v;
                int   ly = jy + dy;
#pragma unroll
                for (int dx = 0; dx < 2; ++dx) {
                    float wx = dx ? wx1 : wx0v;
                    int   lx = jx + dx;
                    float w  = wz * wy * wx;
                    const float* gc = &lds_grid[((ly * 3 + lx) * GD + zi) * NC];
                    o0 += w * (gc[0] * xv0 + gc[1]  * xv1 + gc[2]  * xv2 + gc[3]);
                    o1 += w * (gc[4] * xv0 + gc[5]  * xv1 + gc[6]  * xv2 + gc[7]);
                    o2 += w * (gc[8] * xv0 + gc[9]  * xv1 + gc[10] * xv2 + gc[11]);
                }
            }
        }
        size_t ob = (((size_t)batch * Hn + yg) * Wn + xg) * Cn;
        out[ob + 0] = o0;
        out[ob + 1] = o1;
        out[ob + 2] = o2;
    }
}

extern "C" void kernel_launch(void* const* d_in, const int* in_sizes, int n_in,
                              void* d_out, int out_size, void* d_ws, size_t ws_size,
                              hipStream_t stream) {
    const float* bgrid  = (const float*)d_in[0];
    const float* guide  = (const float*)d_in[1];
    const float* image  = (const float*)d_in[2];
    const float* conv_w = (const float*)d_in[3];
    const float* conv_b = (const float*)d_in[4];
    float* out = (float*)d_out;

    dim3 grd(Wn / TW, Hn / TH, 4);
    fused_conv_bilateral<<<grd, NTHREADS, 0, stream>>>(bgrid, guide, image,
                                                       conv_w, conv_b, out);
}